// BangaloreGCN_83193516524092
// MI455X (gfx1250) — compile-verified
//
#include <hip/hip_runtime.h>

#define BN_EPS 1e-5f

typedef __attribute__((ext_vector_type(2))) float v2f;
typedef __attribute__((ext_vector_type(8))) float v8f;

// ---------------- degree / normalization ----------------

__global__ void k_deg_init(float* __restrict__ deg, int N) {
  int i = blockIdx.x * blockDim.x + threadIdx.x;
  if (i < N) deg[i] = 1.0f;  // self loop contributes 1 to every node's degree
}

__global__ void k_deg_accum(const int* __restrict__ col, float* __restrict__ deg, int E) {
  int e = blockIdx.x * blockDim.x + threadIdx.x;
  if (e < E) atomicAdd(&deg[col[e]], 1.0f);
}

__global__ void k_dinv(float* __restrict__ deg, int N) {
  int i = blockIdx.x * blockDim.x + threadIdx.x;
  if (i < N) { float d = deg[i]; deg[i] = d > 0.0f ? rsqrtf(d) : 0.0f; }
}

// ---------------- f32 WMMA GEMM: C[M,NF] = A[M,K] * B[K,NF] ----------------
// W (B operand) staged once per block in LDS; each wave computes a full 16xNF
// strip with NF/16 independent WMMA accumulator chains (V_WMMA_F32_16X16X4_F32).
// A 16x4 layout (ISA 7.12.2): lanes 0-15 hold K=k,k+1 ; lanes 16-31 hold K=k+2,k+3,
// row M = lane&15. B 4x16 mirrored (row K striped across lanes, N = lane&15).
// C: VGPR r -> M = 16*mt + r + 8*(lane>=16), N = 16*nt + (lane&15).
template <int K, int NF>
__global__ __launch_bounds__(128) void k_gemm_wmma(const float* __restrict__ A,
                                                   const float* __restrict__ B,
                                                   float* __restrict__ C, int M) {
  constexpr int NT = NF / 16;           // n-tiles per strip (4 or 2)
  __shared__ float ldsW[K * NF];        // 32KB (128x64) or 8KB (64x32)

  // cooperative load of the whole weight matrix into LDS (K*NF % 512 == 0)
  for (int idx = threadIdx.x * 4; idx < K * NF; idx += 128 * 4) {
    *(float4*)&ldsW[idx] = *(const float4*)&B[idx];
  }
  __syncthreads();

  const int lane = threadIdx.x & 31;
  const int wave = threadIdx.x >> 5;
  const int ln = lane & 15;
  const int hi = lane >> 4;  // 0 or 1

  const int mtiles = M / 16;
  int mt = blockIdx.x * 4 + wave;       // wave-uniform
  if (mt >= mtiles) return;             // whole wave exits; EXEC stays all-1s for WMMA

  const float* arow = A + (size_t)(mt * 16 + ln) * K + 2 * hi;
  const float* bbase = ldsW + (size_t)(2 * hi) * NF + ln;

  v8f acc[NT];
#pragma unroll
  for (int nt = 0; nt < NT; ++nt) acc[nt] = (v8f){};

#pragma unroll
  for (int k = 0; k < K; k += 4) {
    v2f a;
    a.x = arow[k];
    a.y = arow[k + 1];
#pragma unroll
    for (int nt = 0; nt < NT; ++nt) {
      v2f b;
      b.x = bbase[(size_t)k * NF + nt * 16];        // ds_load
      b.y = bbase[(size_t)(k + 1) * NF + nt * 16];  // ds_load
      acc[nt] = __builtin_amdgcn_wmma_f32_16x16x4_f32(false, a, false, b, (short)0,
                                                      acc[nt], false, false);
    }
  }

  float* crow = C + (size_t)(mt * 16 + hi * 8) * NF + ln;
#pragma unroll
  for (int nt = 0; nt < NT; ++nt) {
#pragma unroll
    for (int r = 0; r < 8; ++r) crow[(size_t)r * NF + nt * 16] = acc[nt][r];
  }
}

// ---------------- self-loop init: dst[i,f] = src[i,f] * dinv[i]^2 (float4) ------

template <int LOGQ>  // LOGQ = log2(F/4)
__global__ void k_selfloop4(const float* __restrict__ src, const float* __restrict__ dinv,
                            float* __restrict__ dst, long long total4) {
  long long t = (long long)blockIdx.x * blockDim.x + threadIdx.x;
  if (t >= total4) return;
  int i = (int)(t >> LOGQ);
  float di = dinv[i];
  float w = di * di;
  float4 v = *(const float4*)(src + t * 4);
  v.x *= w; v.y *= w; v.z *= w; v.w *= w;
  *(float4*)(dst + t * 4) = v;
}

// ---- edge scatter: dst[col, f..f+3] += src[row, f..f+3] * dinv[row]*dinv[col] ----

template <int F, int LOGQ>  // LOGQ = log2(F/4)
__global__ void k_scatter4(const int* __restrict__ row, const int* __restrict__ col,
                           const float* __restrict__ dinv, const float* __restrict__ src,
                           float* __restrict__ dst, long long total4) {
  long long t = (long long)blockIdx.x * blockDim.x + threadIdx.x;
  if (t >= total4) return;
  int e = (int)(t >> LOGQ);
  int q = (int)(t & ((F / 4) - 1));
  int r = row[e];
  int c = col[e];
  float w = dinv[r] * dinv[c];
  float4 v = *(const float4*)(src + (size_t)r * F + q * 4);  // global_load_b128
  float* d = dst + (size_t)c * F + q * 4;
  atomicAdd(d + 0, v.x * w);
  atomicAdd(d + 1, v.y * w);
  atomicAdd(d + 2, v.z * w);
  atomicAdd(d + 3, v.w * w);
}

// ---------------- BN(eval) + ReLU epilogue (adds conv bias first, float4) --------

template <int F>
__global__ void k_bn_relu4(const float* __restrict__ agg, const float* __restrict__ bias,
                           const float* __restrict__ g, const float* __restrict__ be,
                           const float* __restrict__ rm, const float* __restrict__ rv,
                           float* __restrict__ out, long long total4) {
  long long t = (long long)blockIdx.x * blockDim.x + threadIdx.x;
  if (t >= total4) return;
  int f = (int)(t & ((F / 4) - 1)) * 4;
  float4 v = *(const float4*)(agg + t * 4);
  float4 bi = *(const float4*)(bias + f);
  float4 gg = *(const float4*)(g + f);
  float4 bb = *(const float4*)(be + f);
  float4 mm = *(const float4*)(rm + f);
  float4 vv = *(const float4*)(rv + f);
  float4 o;
  o.x = fmaxf((v.x + bi.x - mm.x) * rsqrtf(vv.x + BN_EPS) * gg.x + bb.x, 0.0f);
  o.y = fmaxf((v.y + bi.y - mm.y) * rsqrtf(vv.y + BN_EPS) * gg.y + bb.y, 0.0f);
  o.z = fmaxf((v.z + bi.z - mm.z) * rsqrtf(vv.z + BN_EPS) * gg.z + bb.z, 0.0f);
  o.w = fmaxf((v.w + bi.w - mm.w) * rsqrtf(vv.w + BN_EPS) * gg.w + bb.w, 0.0f);
  *(float4*)(out + t * 4) = o;
}

// ---------------- fused BN + ReLU + final 32->1 Linear ----------------

__global__ void k_final(const float* __restrict__ agg, const float* __restrict__ b2,
                        const float* __restrict__ g2, const float* __restrict__ be2,
                        const float* __restrict__ rm2, const float* __restrict__ rv2,
                        const float* __restrict__ fcW, const float* __restrict__ fcb,
                        float* __restrict__ out, int N) {
  int i = blockIdx.x * blockDim.x + threadIdx.x;
  if (i >= N) return;
  const float* a = agg + (size_t)i * 32;
  float s = 0.0f;
#pragma unroll
  for (int q = 0; q < 8; ++q) {
    float4 v = *(const float4*)(a + q * 4);
    float4 bi = *(const float4*)(b2 + q * 4);
    float4 gg = *(const float4*)(g2 + q * 4);
    float4 bb = *(const float4*)(be2 + q * 4);
    float4 mm = *(const float4*)(rm2 + q * 4);
    float4 vv = *(const float4*)(rv2 + q * 4);
    float4 w = *(const float4*)(fcW + q * 4);
    s += fmaxf((v.x + bi.x - mm.x) * rsqrtf(vv.x + BN_EPS) * gg.x + bb.x, 0.0f) * w.x;
    s += fmaxf((v.y + bi.y - mm.y) * rsqrtf(vv.y + BN_EPS) * gg.y + bb.y, 0.0f) * w.y;
    s += fmaxf((v.z + bi.z - mm.z) * rsqrtf(vv.z + BN_EPS) * gg.z + bb.z, 0.0f) * w.z;
    s += fmaxf((v.w + bi.w - mm.w) * rsqrtf(vv.w + BN_EPS) * gg.w + bb.w, 0.0f) * w.w;
  }
  out[i] = s + fcb[0];
}

// ---------------- host side ----------------

extern "C" void kernel_launch(void* const* d_in, const int* in_sizes, int n_in,
                              void* d_out, int out_size, void* d_ws, size_t ws_size,
                              hipStream_t stream) {
  (void)in_sizes; (void)n_in; (void)out_size; (void)ws_size;

  const float* x   = (const float*)d_in[0];
  const int*   ei  = (const int*)d_in[1];
  const float* W1  = (const float*)d_in[2];
  const float* b1  = (const float*)d_in[3];
  const float* W2  = (const float*)d_in[4];
  const float* b2  = (const float*)d_in[5];
  const float* fcW = (const float*)d_in[6];
  const float* fcb = (const float*)d_in[7];
  const float* g1  = (const float*)d_in[8];
  const float* be1 = (const float*)d_in[9];
  const float* rm1 = (const float*)d_in[10];
  const float* rv1 = (const float*)d_in[11];
  const float* g2  = (const float*)d_in[12];
  const float* be2 = (const float*)d_in[13];
  const float* rm2 = (const float*)d_in[14];
  const float* rv2 = (const float*)d_in[15];

  const int N = 50000;
  const int E = 800000;
  const int* row = ei;      // edge_index[0] : source
  const int* col = ei + E;  // edge_index[1] : target

  // carve workspace (≈38.6 MB total)
  char* ws = (char*)d_ws;
  size_t off = 0;
  auto carve = [&](size_t bytes) -> void* {
    void* p = (void*)(ws + off);
    off += (bytes + 255) & ~(size_t)255;
    return p;
  };
  float* dinv = (float*)carve((size_t)N * 4);        // deg -> dinv (in place)
  float* buf1 = (float*)carve((size_t)N * 64 * 4);   // h1, then a1 (post-BN activations)
  float* buf2 = (float*)carve((size_t)N * 64 * 4);   // agg1
  float* buf3 = (float*)carve((size_t)N * 32 * 4);   // h2
  float* buf4 = (float*)carve((size_t)N * 32 * 4);   // agg2

  // 1) degrees + symmetric normalization
  k_deg_init<<<(N + 255) / 256, 256, 0, stream>>>(dinv, N);
  k_deg_accum<<<(E + 255) / 256, 256, 0, stream>>>(col, dinv, E);
  k_dinv<<<(N + 255) / 256, 256, 0, stream>>>(dinv, N);

  // 2) h1 = x @ W1  (WMMA f32; one wave per 16x64 strip, W1 in LDS)
  {
    int mtiles = N / 16;  // 3125
    k_gemm_wmma<128, 64><<<(mtiles + 3) / 4, 128, 0, stream>>>(x, W1, buf1, N);
  }

  // 3) aggregate layer 1: self loops as init, edges via f32 atomics (L2-resident)
  long long t1 = (long long)N * 16;  // N*64/4 float4 elements
  k_selfloop4<4><<<(unsigned)((t1 + 255) / 256), 256, 0, stream>>>(buf1, dinv, buf2, t1);
  long long s1 = (long long)E * 16;  // E*64/4
  k_scatter4<64, 4><<<(unsigned)((s1 + 255) / 256), 256, 0, stream>>>(row, col, dinv,
                                                                      buf1, buf2, s1);
  // 4) +b1, BN1, ReLU -> a1 (reuse buf1)
  k_bn_relu4<64><<<(unsigned)((t1 + 255) / 256), 256, 0, stream>>>(buf2, b1, g1, be1,
                                                                   rm1, rv1, buf1, t1);

  // 5) h2 = a1 @ W2 (WMMA f32; one wave per 16x32 strip, W2 in LDS)
  {
    int mtiles = N / 16;
    k_gemm_wmma<64, 32><<<(mtiles + 3) / 4, 128, 0, stream>>>(buf1, W2, buf3, N);
  }

  // 6) aggregate layer 2
  long long t2 = (long long)N * 8;  // N*32/4
  k_selfloop4<3><<<(unsigned)((t2 + 255) / 256), 256, 0, stream>>>(buf3, dinv, buf4, t2);
  long long s2 = (long long)E * 8;  // E*32/4
  k_scatter4<32, 3><<<(unsigned)((s2 + 255) / 256), 256, 0, stream>>>(row, col, dinv,
                                                                      buf3, buf4, s2);

  // 7) +b2, BN2, ReLU, FC -> out [N,1]
  k_final<<<(N + 255) / 256, 256, 0, stream>>>(buf4, b2, g2, be2, rm2, rv2, fcW, fcb,
                                               (float*)d_out, N);
}